// OrthonormalWaveletBlock1D_47510928228979
// MI455X (gfx1250) — compile-verified
//
#include <hip/hip_runtime.h>

typedef __attribute__((ext_vector_type(2))) float v2f;
typedef __attribute__((ext_vector_type(4))) float v4f;
typedef __attribute__((ext_vector_type(8))) float v8f;
typedef __attribute__((ext_vector_type(4))) int   v4i;

#define KSZ       10
#define TILE_OUT  1024
#define LDS_N     (2 * TILE_OUT + 8)   // 2056 padded input samples per WG tile
#define NCHUNK    6                    // K = 24 taps (band-extended), 6 x K=4 WMMA
#define FTAB_N    40                   // filt_f[t] table, t in [-14, 25], zero outside [0,9]

// gfx1250 async global->LDS path, compile-safe probe
#if defined(__has_builtin)
#if __has_builtin(__builtin_amdgcn_global_load_async_to_lds_b128) && \
    __has_builtin(__builtin_amdgcn_s_wait_asynccnt)
#define USE_ASYNC_LDS 1
#endif
#endif
#ifndef USE_ASYNC_LDS
#define USE_ASYNC_LDS 0
#endif

#if USE_ASYNC_LDS
// exact param types per hipcc diagnostic: arg0 is (v4i __device__ *), arg1 LDS side
typedef __attribute__((address_space(1))) v4i* gas_v4i_ptr;
typedef __attribute__((address_space(3))) v4i* las_v4i_ptr;
#endif

__global__ __launch_bounds__(256)
void wavelet_level_kernel(const float* __restrict__ x,
                          const float* __restrict__ hker,
                          float* __restrict__ a_out,
                          float* __restrict__ d_out,
                          int L, int Lout)
{
    __shared__ float lds[LDS_N];
    __shared__ float ftab[2 * FTAB_N];

    const int row   = blockIdx.y;            // (b*64 + c), 0..511
    const int obase = blockIdx.x * TILE_OUT; // first output position of this WG
    const int tid   = threadIdx.x;

    const float* xrow = x + (long)row * (long)L;

    // ---- build zero-padded filter table in LDS (no register-indexed selects) ----
    // ftab[f*40 + (t+14)] = f==0 ? h[t] : h[9-t]*(-1)^t   for t in [0,9], else 0
    if (tid < 2 * FTAB_N) {
        int t = (tid % FTAB_N) - 14;
        float v = 0.0f;
        if (t >= 0 && t <= 9) {
            v = (tid < FTAB_N) ? hker[t] : ((t & 1) ? -hker[9 - t] : hker[9 - t]);
        }
        ftab[tid] = v;
    }

    // ---- stage padded window into LDS: lds[i] == xp[2*obase - 4 + i] ----
    const int q0  = 2 * obase - 4;
    const int iLo = (q0 < 0) ? -q0 : 0;                    // first in-range i (0 or 4)
    const int iHi = (L - q0 < LDS_N) ? (L - q0) : LDS_N;   // one past last in-range i
    const int n4  = (iHi - iLo) >> 2;                      // 16B chunks (exact)

#if USE_ASYNC_LDS
    for (int t4 = tid; t4 < n4; t4 += 256) {
        int i = iLo + 4 * t4;
        __builtin_amdgcn_global_load_async_to_lds_b128(
            (gas_v4i_ptr)(xrow + q0 + i),
            (las_v4i_ptr)&lds[i], 0, 0);
    }
    __builtin_amdgcn_s_wait_asynccnt(0);
#else
    for (int t4 = tid; t4 < n4; t4 += 256) {
        int i = iLo + 4 * t4;
        *(v4f*)&lds[i] = *(const v4f*)&xrow[q0 + i];
    }
#endif
    __syncthreads();

    // ---- antireflect edge patch (reflected samples are inside staged interior) ----
    if (q0 < 0 && tid < iLo) {
        // xp[q] = 2*x[0] - x[-q] ; x[0] at lds[iLo]
        lds[tid] = 2.0f * lds[iLo] - lds[2 * iLo - tid];
    }
    if (iHi < LDS_N && tid >= 8 && tid < 8 + (LDS_N - iHi)) {
        int i = iHi + (tid - 8);
        // xp[q>=L] = 2*x[L-1] - x[2L-2-q] ; x[L-1] at lds[iHi-1]
        lds[i] = 2.0f * lds[iHi - 1] - lds[2 * (iHi - 1) - i];
    }
    __syncthreads();

    // ---- per-lane WMMA fragment coordinates (wave32) ----
    const int lane = tid & 31;
    const int wave = tid >> 5;
    const int kb   = (lane >> 4) << 1;  // K base for this lane half: 0 or 2
    const int n    = lane & 15;         // B/D column
    const int f    = n >> 3;            // 0 = scaling (a), 1 = wavelet (d)
    const int s    = n & 7;             // output sub-shift within row-block
    const int m    = lane & 15;         // A row

    // ---- B fragments from LDS table: B[k', f*8+s] = filt_f[k' - 2s] ----
    const int tb = f * FTAB_N + 14 - 2 * s;   // ftab base for this lane
    v2f bfr[NCHUNK];
#pragma unroll
    for (int j = 0; j < NCHUNK; ++j) {
        int kp = 4 * j + kb;
        bfr[j].x = ftab[tb + kp];
        bfr[j].y = ftab[tb + kp + 1];
    }

    // ---- A fragments from LDS: A[m,k'] = xp[2*wbase + 16m - 4 + k'] ----
    const int abase = 256 * wave + 16 * m;

    v8f c = {};
#pragma unroll
    for (int j = 0; j < NCHUNK; ++j) {
        v2f a;
        a.x = lds[abase + 4 * j + kb];
        a.y = lds[abase + 4 * j + kb + 1];
        // D = A*B + C ; chained f32 WMMA accumulation
        c = __builtin_amdgcn_wmma_f32_16x16x4_f32(false, a, false, bfr[j],
                                                  (short)0, c, false, false);
    }
    __syncthreads();   // all LDS window reads done; safe to reuse lds as store stage

    // ---- stage D tile into LDS: per-wave 256 floats [a:128][d:128] ----
    const int mhi   = (lane >> 4) * 8;
    const int stage = wave * 256 + f * 128;
#pragma unroll
    for (int v = 0; v < 8; ++v) {
        int mm = v + mhi;                 // D row
        lds[stage + 8 * mm + s] = c[v];   // local output offset = 8*M + s
    }
    __syncthreads();

    // ---- coalesced stores: each lane one 16B chunk per output array ----
    const int  wbase = obase + wave * 128;
    const long rb    = (long)row * (long)Lout + wbase;
    v4f av = *(const v4f*)&lds[wave * 256 + 4 * lane];
    v4f dv = *(const v4f*)&lds[wave * 256 + 128 + 4 * lane];
    *(v4f*)&a_out[rb + 4 * lane] = av;
    *(v4f*)&d_out[rb + 4 * lane] = dv;
}

extern "C" void kernel_launch(void* const* d_in, const int* in_sizes, int n_in,
                              void* d_out, int out_size, void* d_ws, size_t ws_size,
                              hipStream_t stream) {
    (void)in_sizes; (void)n_in; (void)out_size; (void)d_ws; (void)ws_size;

    const float* signal = (const float*)d_in[0];   // [8,64,65536] f32
    const float* hker   = (const float*)d_in[1];   // [1,1,10]     f32
    float*       out    = (float*)d_out;           // a1..a4 then d1..d4, flat

    const int ROWS   = 8 * 64;   // 512 depthwise rows
    const int L0     = 65536;
    const int LEVELS = 4;

    // output offsets: approximations first, then details (reference tuple order)
    long aoff[4], doff[4];
    long off = 0;
    for (int l = 0; l < LEVELS; ++l) { aoff[l] = off; off += (long)ROWS * (long)(L0 >> (l + 1)); }
    for (int l = 0; l < LEVELS; ++l) { doff[l] = off; off += (long)ROWS * (long)(L0 >> (l + 1)); }

    const float* src = signal;
    for (int l = 0; l < LEVELS; ++l) {
        int Ll = L0 >> l;       // input length this level
        int Lo = Ll >> 1;       // output length this level (multiple of TILE_OUT at all levels)
        dim3 grid(Lo / TILE_OUT, ROWS);
        wavelet_level_kernel<<<grid, 256, 0, stream>>>(
            src, hker, out + aoff[l], out + doff[l], Ll, Lo);
        src = out + aoff[l];    // approximation feeds next level
    }
}